// GCN_5531917877789
// MI455X (gfx1250) — compile-verified
//
#include <hip/hip_runtime.h>
#include <hip/hip_bf16.h>

typedef float v2f __attribute__((ext_vector_type(2)));
typedef float v8f __attribute__((ext_vector_type(8)));

// ---------------------------------------------------------------------------
// Utility kernels
// ---------------------------------------------------------------------------
__global__ void gcn_zero_f(float* __restrict__ p, int n) {
    int t = blockIdx.x * blockDim.x + threadIdx.x;
    if (t < n) p[t] = 0.0f;
}

__global__ void gcn_copy4(float4* __restrict__ dst, const float4* __restrict__ src, int n4) {
    int t = blockIdx.x * blockDim.x + threadIdx.x;
    if (t < n4) dst[t] = src[t];
}

__global__ void gcn_deg(const int* __restrict__ se, const int* __restrict__ re,
                        float* __restrict__ degS, float* __restrict__ degR, int E) {
    int t = blockIdx.x * blockDim.x + threadIdx.x;
    if (t < E) {
        atomicAdd(&degS[se[t]], 1.0f);
        atomicAdd(&degR[re[t]], 1.0f);
    }
}

// dst[r[e]][f..f+3] += src[s[e]][f..f+3]  (one thread per edge-feature-quad:
// one global_load_b128 + 4 global_atomic_add_f32, indices read once per quad)
template <int F>
__global__ void gcn_scatter(const int* __restrict__ se, const int* __restrict__ re,
                            const float* __restrict__ src, float* __restrict__ dst, int E) {
    constexpr int FQ = F / 4;
    long long t = (long long)blockIdx.x * blockDim.x + threadIdx.x;
    if (t >= (long long)E * FQ) return;
    int e = (int)(t / FQ);
    int f = (int)(t % FQ) * 4;
    int s = se[e];
    int r = re[e];
    float4 v = *reinterpret_cast<const float4*>(&src[(size_t)s * F + f]);
    float* d = &dst[(size_t)r * F + f];
    atomicAdd(d + 0, v.x);
    atomicAdd(d + 1, v.y);
    atomicAdd(d + 2, v.z);
    atomicAdd(d + 3, v.w);
}

// out[i][f] = in[i][f] * rsqrt(deg_r[i] (+1 | clamped to >=1))
template <int F, bool ADDONE>
__global__ void gcn_scale(const float* __restrict__ in, const float* __restrict__ degR,
                          float* __restrict__ out, int N) {
    long long t = (long long)blockIdx.x * blockDim.x + threadIdx.x;
    if (t >= (long long)N * F) return;
    int i = (int)(t >> (F == 64 ? 6 : 4));
    float d = degR[i];
    float rs = ADDONE ? rsqrtf(d + 1.0f) : rsqrtf(fmaxf(d, 1.0f));
    out[t] = in[t] * rs;
}

// ---------------------------------------------------------------------------
// WMMA GEMM: out[i][n] = act(in[i][:] @ W[:][n] + b[n]) * rsqrt(deg_s[i] ...)
// in: N x 64 (f32), W: 64 x NOUT (f32), out: N x NOUT
// Block = 128 threads = 4 waves; each wave -> one 16-row tile, all NOUT cols.
// V_WMMA_F32_16X16X4_F32 fragment layouts (ISA 7.12.2):
//   A 16x4:  lane L holds row M=L&15, K = k0 + 2*(L>>4) + {0,1}
//   B 4x16:  lane L holds col N=L&15, K = k0 + 2*(L>>4) + {0,1}
//   C/D:     lane L holds col N=L&15, vgpr j -> row M = j + 8*(L>>4)
// LDS: A staged row-major pitch 68, W staged TRANSPOSED (Wt[n][k]) pitch 68.
//   -> each A/B fragment is one contiguous 8B ds_load_b64, and bank index is
//      (4*lane + k) % 64: conflict-free across both half-waves.
// ---------------------------------------------------------------------------
template <int NOUT, bool TANH, bool ADDONE>
__global__ __launch_bounds__(128) void gcn_gemm(const float* __restrict__ in,
                                                const float* __restrict__ W,
                                                const float* __restrict__ bias,
                                                const float* __restrict__ degS,
                                                float* __restrict__ out, int N) {
    constexpr int K = 64;
    constexpr int CT = NOUT / 16;
    constexpr int PITCH = 68;  // 68*4B: 8B-aligned rows, bank stride 4

    __shared__ float Wt[NOUT * PITCH];  // Wt[n*PITCH + k] = W[k][n]
    __shared__ float Bl[NOUT];
    __shared__ float Al[64 * PITCH];    // Al[row*PITCH + k], 64 rows per block

    const int tid = threadIdx.x;

    // Stage W transposed: coalesced float4 global reads, scattered LDS writes.
    const float4* Wg4 = reinterpret_cast<const float4*>(W);
    for (int i = tid; i < K * NOUT / 4; i += 128) {
        int k = i / (NOUT / 4);
        int n = (i % (NOUT / 4)) * 4;
        float4 w = Wg4[i];
        Wt[(n + 0) * PITCH + k] = w.x;
        Wt[(n + 1) * PITCH + k] = w.y;
        Wt[(n + 2) * PITCH + k] = w.z;
        Wt[(n + 3) * PITCH + k] = w.w;
    }
    if (tid < NOUT) Bl[tid] = bias[tid];

    // Stage this block's 64 rows of A: coalesced float4 reads, float4 LDS writes.
    const int rowBase = blockIdx.x * 64;
    const float4* Ag4 = reinterpret_cast<const float4*>(in);
    for (int i = tid; i < 64 * (K / 4); i += 128) {
        int row = rowBase + i / (K / 4);
        int c4 = (i % (K / 4)) * 4;
        float4 v = (row < N) ? Ag4[(size_t)row * (K / 4) + (c4 >> 2)]
                             : make_float4(0.f, 0.f, 0.f, 0.f);
        *reinterpret_cast<float4*>(&Al[(i / (K / 4)) * PITCH + c4]) = v;
    }
    __syncthreads();

    const int wave = tid >> 5;
    const int lane = tid & 31;
    const int tileRow = rowBase + wave * 16;
    if (tileRow >= N) return;  // wave-uniform: EXEC all-ones inside

    const int lm = lane & 15;        // A row-in-tile / B,D col-in-tile
    const int kh = (lane >> 4) * 2;  // K sub-offset for upper half-wave
    const int mh = (lane >> 4) * 8;  // D row offset for upper half-wave

    v8f acc[CT] = {};
    const float* Arow = &Al[(wave * 16 + lm) * PITCH + kh];

#pragma unroll
    for (int k0 = 0; k0 < K; k0 += 4) {
        v2f a = *reinterpret_cast<const v2f*>(&Arow[k0]);
#pragma unroll
        for (int c = 0; c < CT; ++c) {
            v2f b = *reinterpret_cast<const v2f*>(&Wt[(c * 16 + lm) * PITCH + k0 + kh]);
            acc[c] = __builtin_amdgcn_wmma_f32_16x16x4_f32(
                false, a, false, b, (short)0, acc[c], false, false);
        }
    }

    // Row scales (pre-normalization by sender degree), one per D vgpr slot.
    float rs[8];
#pragma unroll
    for (int j = 0; j < 8; ++j) {
        int row = tileRow + mh + j;
        float d = (row < N) ? degS[row] : 1.0f;
        rs[j] = ADDONE ? rsqrtf(d + 1.0f) : rsqrtf(fmaxf(d, 1.0f));
    }

#pragma unroll
    for (int c = 0; c < CT; ++c) {
        int n = c * 16 + lm;
        float bv = Bl[n];
#pragma unroll
        for (int j = 0; j < 8; ++j) {
            int row = tileRow + mh + j;
            if (row < N) {
                float v = acc[c][j] + bv;
                if (TANH) v = tanhf(v);
                out[(size_t)row * NOUT + n] = v * rs[j];
            }
        }
    }
}

// ---------------------------------------------------------------------------
// Launch: full 3-layer GCN pipeline
// ---------------------------------------------------------------------------
extern "C" void kernel_launch(void* const* d_in, const int* in_sizes, int n_in,
                              void* d_out, int out_size, void* d_ws, size_t ws_size,
                              hipStream_t stream) {
    const float* nodes = (const float*)d_in[0];
    const int* senders = (const int*)d_in[1];
    const int* receivers = (const int*)d_in[2];
    const float* W0 = (const float*)d_in[3];
    const float* b0 = (const float*)d_in[4];
    const float* W1 = (const float*)d_in[5];
    const float* b1 = (const float*)d_in[6];
    const float* W2 = (const float*)d_in[7];
    const float* b2 = (const float*)d_in[8];
    float* out = (float*)d_out;

    const int N = in_sizes[0] / 64;
    const int E = in_sizes[1];

    float* X = (float*)d_ws;                // N x 64
    float* Y = X + (size_t)N * 64;          // N x 64
    float* Z = Y + (size_t)N * 64;          // N x 16
    float* degS = Z + (size_t)N * 16;       // N
    float* degR = degS + N;                 // N

    const int zn = N * 16 + 2 * N;  // Z + degS + degR are contiguous
    gcn_zero_f<<<(zn + 255) / 256, 256, 0, stream>>>(Z, zn);
    gcn_deg<<<(E + 255) / 256, 256, 0, stream>>>(senders, receivers, degS, degR, E);

    const int tiles = (N + 15) / 16;
    const int gblocks = (tiles + 3) / 4;
    const int cp4 = N * 16;  // N*64/4 float4s
    const int g64 = (int)(((long long)E * 16 + 255) / 256);  // edge-quads, F=64
    const int g16 = (int)(((long long)E * 4 + 255) / 256);   // edge-quads, F=16
    const int s64 = (int)(((long long)N * 64 + 255) / 256);
    const int s16 = (int)(((long long)N * 16 + 255) / 256);

    // ---- Layer 0: tanh, self edges ----
    gcn_gemm<64, true, true><<<gblocks, 128, 0, stream>>>(nodes, W0, b0, degS, X, N);
    gcn_copy4<<<(cp4 + 255) / 256, 256, 0, stream>>>((float4*)Y, (const float4*)X, cp4);  // self edge
    gcn_scatter<64><<<g64, 256, 0, stream>>>(senders, receivers, X, Y, E);
    gcn_scale<64, true><<<s64, 256, 0, stream>>>(Y, degR, X, N);

    // ---- Layer 1: tanh, self edges ----
    gcn_gemm<64, true, true><<<gblocks, 128, 0, stream>>>(X, W1, b1, degS, Y, N);
    gcn_copy4<<<(cp4 + 255) / 256, 256, 0, stream>>>((float4*)X, (const float4*)Y, cp4);  // self edge
    gcn_scatter<64><<<g64, 256, 0, stream>>>(senders, receivers, Y, X, E);
    gcn_scale<64, true><<<s64, 256, 0, stream>>>(X, degR, Y, N);

    // ---- Layer 2: linear, no self edges ----
    gcn_gemm<16, false, false><<<gblocks, 128, 0, stream>>>(Y, W2, b2, degS, X, N);
    gcn_scatter<16><<<g16, 256, 0, stream>>>(senders, receivers, X, Z, E);
    gcn_scale<16, false><<<s16, 256, 0, stream>>>(Z, degR, out, N);
}